// MultiheadSelfAttention_29489245454525
// MI455X (gfx1250) — compile-verified
//
#include <hip/hip_runtime.h>

typedef float v2f __attribute__((ext_vector_type(2)));
typedef float v4f __attribute__((ext_vector_type(4)));
typedef float v8f __attribute__((ext_vector_type(8)));

#define DIMC   1024
#define NHEADS 16
#define HD     64
#define SEQ    2048
#define BATCH  2
#define QK_SCALE 0.125f   // 64^-0.5
#define NEGINF  (-3.0e38f)

__device__ __forceinline__ v8f wmma4(v2f a, v2f b, v8f c) {
    // D = A(16x4, f32) x B(4x16, f32) + C(16x16, f32)
    return __builtin_amdgcn_wmma_f32_16x16x4_f32(
        /*neg_a=*/false, a, /*neg_b=*/false, b,
        /*c_mod=*/(short)0, c, /*reuse_a=*/false, /*reuse_b=*/false);
}

// C[M,Nout] = A[M,K] * W[Nout,K]^T + bias[Nout]
// One wave computes a 32x32 tile (2x2 register blocking): 4 B64 loads feed
// 4 WMMAs per K-step of 4 -> 1 load per v_wmma, half the L2 traffic of 16x16.
__global__ __launch_bounds__(256)
void gemm_wmma(const float* __restrict__ A, const float* __restrict__ W,
               const float* __restrict__ bias, float* __restrict__ C,
               int M, int Nout, int K) {
    const int lane  = threadIdx.x & 31;
    const int wave  = blockIdx.x * 8 + (threadIdx.x >> 5);
    const int tilesN = Nout >> 5;          // 32-wide tiles
    const int tM = (wave / tilesN) << 5;
    const int tN = (wave % tilesN) << 5;
    if (tM >= M) return;

    const int l15 = lane & 15;
    const int kk2 = (lane & 16) >> 3;      // K sub-offset: 0 (lanes 0-15) / 2 (lanes 16-31)

    const float* aPtr0 = A + (size_t)(tM + l15) * K + kk2;
    const float* aPtr1 = aPtr0 + (size_t)16 * K;
    const float* wPtr0 = W + (size_t)(tN + l15) * K + kk2;
    const float* wPtr1 = wPtr0 + (size_t)16 * K;

    v8f acc00 = {}, acc01 = {}, acc10 = {}, acc11 = {};
    for (int k = 0; k < K; k += 8) {
#pragma unroll
        for (int kk = 0; kk < 8; kk += 4) {
            v2f a0 = *(const v2f*)(aPtr0 + k + kk);
            v2f a1 = *(const v2f*)(aPtr1 + k + kk);
            v2f b0 = *(const v2f*)(wPtr0 + k + kk);
            v2f b1 = *(const v2f*)(wPtr1 + k + kk);
            acc00 = wmma4(a0, b0, acc00);
            acc01 = wmma4(a0, b1, acc01);
            acc10 = wmma4(a1, b0, acc10);
            acc11 = wmma4(a1, b1, acc11);
        }
    }

    const int c0 = tN + l15, c1 = c0 + 16;
    const float bv0 = bias[c0], bv1 = bias[c1];
    const int r0 = tM + ((lane & 16) >> 1);   // +8 for upper half-wave
#pragma unroll
    for (int i = 0; i < 8; ++i) {
        C[(size_t)(r0 + i)      * Nout + c0] = acc00[i] + bv0;
        C[(size_t)(r0 + i)      * Nout + c1] = acc01[i] + bv1;
        C[(size_t)(r0 + 16 + i) * Nout + c0] = acc10[i] + bv0;
        C[(size_t)(r0 + 16 + i) * Nout + c1] = acc11[i] + bv1;
    }
}

// Flash-attention over one (b, h, 16-query) tile per wave.
// qkv laid out [b, n, 3*DIM] with q at col 0, k at +1024, v at +2048.
// Computes S^T = K*Q^T (softmax rows map to lanes), then O^T = V^T * P^T.
__global__ __launch_bounds__(256)
void attn_wmma(const float* __restrict__ qkv, const float* __restrict__ bias,
               const unsigned char* __restrict__ kpm, float* __restrict__ obuf) {
    const int lane  = threadIdx.x & 31;
    const int wave  = blockIdx.x * 8 + (threadIdx.x >> 5);
    const int qTile = wave & 127;          // SEQ/16 = 128
    const int h     = (wave >> 7) & 15;
    const int b     = wave >> 11;
    const int qBase = qTile << 4;

    const int l15  = lane & 15;
    const int hOff = (lane & 16) >> 1;     // 0 / 8 (row offset in C/D layout)
    const int kk2  = (lane & 16) >> 3;     // 0 / 2 (K offset in A/B layout)
    const bool lo  = (lane < 16);

    const int ld = 3 * DIMC;               // 3072
    const float* qkvB = qkv + (size_t)b * SEQ * ld;

    // Preload Q^T B-frags: B[dd, q] = Q[qBase+q, h*64 + 4c + dd]
    const float* qPtr = qkvB + (size_t)(qBase + l15) * ld + h * HD + kk2;
    v2f qf[16];
#pragma unroll
    for (int c = 0; c < 16; ++c) qf[c] = *(const v2f*)(qPtr + 4 * c);

    v8f o[4] = {};                         // O^T tiles (d-chunks of 16), lane = q
    float mOld = NEGINF, lSum = 0.f;

    const float* kBasePtr = qkvB + DIMC + h * HD;
    const float* vBasePtr = qkvB + 2 * DIMC + h * HD;
    const float* biasRow  = bias + ((size_t)h * SEQ + (qBase + l15)) * SEQ;
    const unsigned char* maskRow = kpm + (size_t)b * SEQ;

    for (int kb = 0; kb < SEQ; kb += 16) {
        // ---- S^T = K_block * Q^T : A[k_local, dd] = K[kb+k_local, d] ----
        const float* kPtr = kBasePtr + (size_t)(kb + l15) * ld + kk2;
        v8f s = {};
#pragma unroll
        for (int c = 0; c < 16; ++c) {
            v2f af = *(const v2f*)(kPtr + 4 * c);
            s = wmma4(af, qf[c], s);
        }

        // s[i]: key k = kb + i + hOff, query q = qBase + l15
        const float* bp = biasRow + kb + hOff;
        v4f b0 = *(const v4f*)(bp);
        v4f b1 = *(const v4f*)(bp + 4);
        unsigned long long mw = *(const unsigned long long*)(maskRow + kb + hOff);

        float sv[8];
#pragma unroll
        for (int i = 0; i < 8; ++i) {
            float bvv = (i < 4) ? b0[i] : b1[i - 4];
            float t = s[i] * QK_SCALE + bvv;
            if ((mw >> (8 * i)) & 0xFFull) t = NEGINF;
            sv[i] = t;
        }

        // ---- online softmax: row (over k) = 8 regs + partner lane (lane^16) ----
        float mx = sv[0];
#pragma unroll
        for (int i = 1; i < 8; ++i) mx = fmaxf(mx, sv[i]);
        mx = fmaxf(mx, __shfl_xor(mx, 16, 32));
        float mNew = fmaxf(mOld, mx);
        float corr = __expf(mOld - mNew);
        float ps = 0.f;
#pragma unroll
        for (int i = 0; i < 8; ++i) { sv[i] = __expf(sv[i] - mNew); ps += sv[i]; }
        ps += __shfl_xor(ps, 16, 32);
        lSum = lSum * corr + ps;
        mOld = mNew;
#pragma unroll
        for (int t = 0; t < 4; ++t)
#pragma unroll
            for (int i = 0; i < 8; ++i) o[t][i] *= corr;

        // ---- B-frags of P^T from C-layout sv[] via shfl_xor(16) ----
        float x0 = __shfl_xor(sv[0], 16, 32), x1 = __shfl_xor(sv[1], 16, 32);
        float x2 = __shfl_xor(sv[2], 16, 32), x3 = __shfl_xor(sv[3], 16, 32);
        float x4 = __shfl_xor(sv[4], 16, 32), x5 = __shfl_xor(sv[5], 16, 32);
        float x6 = __shfl_xor(sv[6], 16, 32), x7 = __shfl_xor(sv[7], 16, 32);
        v2f pbs[4];
        pbs[0] = (v2f){ lo ? sv[0] : x2, lo ? sv[1] : x3 };
        pbs[1] = (v2f){ lo ? sv[4] : x6, lo ? sv[5] : x7 };
        pbs[2] = (v2f){ lo ? x0 : sv[2], lo ? x1 : sv[3] };
        pbs[3] = (v2f){ lo ? x4 : sv[6], lo ? x5 : sv[7] };

        // ---- O^T += V^T * P^T : A[d_local, kk] = V[kb+4c+kk, 16t+d_local] ----
#pragma unroll
        for (int c = 0; c < 4; ++c) {
            const float* vc = vBasePtr + (size_t)(kb + 4 * c + kk2) * ld + l15;
#pragma unroll
            for (int t = 0; t < 4; ++t) {
                v2f a = (v2f){ vc[16 * t], vc[ld + 16 * t] };
                o[t] = wmma4(a, pbs[c], o[t]);
            }
        }
    }

    // ---- normalize + store: element (d = 16t+i+hOff, q = l15) ----
    const float inv = 1.0f / lSum;
    float* orow = obuf + (size_t)(b * SEQ + qBase + l15) * DIMC + h * HD + hOff;
#pragma unroll
    for (int t = 0; t < 4; ++t)
#pragma unroll
        for (int i = 0; i < 8; ++i)
            orow[16 * t + i] = o[t][i] * inv;
}

extern "C" void kernel_launch(void* const* d_in, const int* in_sizes, int n_in,
                              void* d_out, int out_size, void* d_ws, size_t ws_size,
                              hipStream_t stream) {
    const float* x         = (const float*)d_in[0];
    const float* attn_bias = (const float*)d_in[1];
    const unsigned char* kpm = (const unsigned char*)d_in[2];
    const float* qkv_w     = (const float*)d_in[3];
    const float* qkv_b     = (const float*)d_in[4];
    const float* proj_w    = (const float*)d_in[5];
    const float* proj_b    = (const float*)d_in[6];
    float* out = (float*)d_out;

    float* qkvBuf = (float*)d_ws;                          // 4096 x 3072 fp32 (48 MB)
    float* oBuf   = qkvBuf + (size_t)4096 * 3072;          // 4096 x 1024 fp32 (16 MB)

    // QKV projection: M=4096, Nout=3072, K=1024 -> 12288 32x32 tiles / 8 waves per block
    gemm_wmma<<<1536, 256, 0, stream>>>(x, qkv_w, qkv_b, qkvBuf, 4096, 3072, 1024);
    // Attention: B*H*(N/16) = 4096 waves -> 512 blocks
    attn_wmma<<<512, 256, 0, stream>>>(qkvBuf, attn_bias, kpm, oBuf);
    // Output projection: M=4096, Nout=1024, K=1024 -> 4096 32x32 tiles
    gemm_wmma<<<512, 256, 0, stream>>>(oBuf, proj_w, proj_b, out, 4096, 1024, 1024);
}